// Net_68710886801700
// MI455X (gfx1250) — compile-verified
//
#include <hip/hip_runtime.h>
#include <hip/hip_bf16.h>

// ---------------------------------------------------------------------------
// GCN (2x GCNConv -> relu(concat) -> GCNConv -> log_softmax) for MI455X.
// fp32 WMMA (V_WMMA_F32_16X16X4_F32) for the dense GEMMs; L2-resident float4
// gathers + global_atomic_add_f32 for the edge scatters (tables fit in 192MB L2).
// ---------------------------------------------------------------------------

typedef float v2f __attribute__((ext_vector_type(2)));
typedef float v8f __attribute__((ext_vector_type(8)));

#define DEV_D   512
#define DEV_H   128
#define DEV_H2  256
#define DEV_C   32

// ---------------- init: deg=1 (self loop), acc=0 ---------------------------
__global__ void init_kernel(float* __restrict__ deg1, float* __restrict__ deg2,
                            float* __restrict__ acc, long long nAcc, int Nn) {
    long long i = (long long)blockIdx.x * blockDim.x + threadIdx.x;
    if (i < Nn) { deg1[i] = 1.0f; deg2[i] = 1.0f; }
    if (i < nAcc) acc[i] = 0.0f;
}

__global__ void zero_kernel(float* __restrict__ p, long long n) {
    long long i = (long long)blockIdx.x * blockDim.x + threadIdx.x;
    if (i < n) p[i] = 0.0f;
}

// ---------------- degree count / rsqrt -------------------------------------
__global__ void deg_kernel(float* __restrict__ deg, const int* __restrict__ dst, int E) {
    int e = blockIdx.x * blockDim.x + threadIdx.x;
    if (e < E) atomicAdd(&deg[dst[e]], 1.0f);
}

__global__ void dinv_kernel(float* __restrict__ deg, int Nn) {
    int i = blockIdx.x * blockDim.x + threadIdx.x;
    if (i < Nn) {
        float d = deg[i];
        deg[i] = (d > 0.0f) ? rsqrtf(d) : 0.0f;
    }
}

// ---------------- fused GEMM: T[N,256] = X[N,512] @ [W1 | W2] --------------
// One wave computes a 16-row x 128-col half (half=0 -> W1, half=1 -> W2):
// 8 WMMA accumulator tiles, K stepped by 4 (f32 WMMA shape 16x16x4).
__global__ __launch_bounds__(256)
void gemm_cat_kernel(const float* __restrict__ X, const float* __restrict__ W1,
                     const float* __restrict__ W2, float* __restrict__ T, int M) {
    const int lane = threadIdx.x & 31;
    const int wave = blockIdx.x * (blockDim.x >> 5) + (threadIdx.x >> 5);
    const int rowTile = wave >> 1;
    const int half    = wave & 1;
    if (rowTile >= (M >> 4)) return;

    const int col = lane & 15;      // B/C column within 16-wide tile
    const int hi  = lane >> 4;      // half-wave select (K offset / M offset)
    const int row0 = rowTile * 16;

    const float* __restrict__ W = half ? W2 : W1;
    // A fragment: row = lane&15, K = k0 + 2*hi + {0,1}  (contiguous float2)
    const float* aPtr = X + (size_t)(row0 + col) * DEV_D + 2 * hi;

    v8f c[8] = {};   // zero-initialized accumulators

    for (int k0 = 0; k0 < DEV_D; k0 += 4) {
        v2f a = *(const v2f*)(aPtr + k0);
        const float* wk = W + (size_t)(k0 + 2 * hi) * DEV_H;  // B rows k, k+1
#pragma unroll
        for (int t = 0; t < 8; ++t) {
            v2f b;
            b.x = wk[t * 16 + col];
            b.y = wk[DEV_H + t * 16 + col];
            c[t] = __builtin_amdgcn_wmma_f32_16x16x4_f32(
                false, a, false, b, (short)0, c[t], false, false);
        }
    }

    // C/D layout: VGPR r holds M = r + 8*hi, N = lane&15
    float* outBase = T + (size_t)row0 * DEV_H2 + half * DEV_H;
#pragma unroll
    for (int t = 0; t < 8; ++t)
#pragma unroll
        for (int r = 0; r < 8; ++r)
            outBase[(size_t)(r + 8 * hi) * DEV_H2 + t * 16 + col] = c[t][r];
}

// ---------------- GEMM3: T3[N,32] = H[N,256] @ W3[256,32] ------------------
__global__ __launch_bounds__(256)
void gemm3_kernel(const float* __restrict__ Hmat, const float* __restrict__ W3,
                  float* __restrict__ T3, int M) {
    const int lane = threadIdx.x & 31;
    const int wave = blockIdx.x * (blockDim.x >> 5) + (threadIdx.x >> 5);
    if (wave >= (M >> 4)) return;

    const int col = lane & 15;
    const int hi  = lane >> 4;
    const int row0 = wave * 16;
    const float* aPtr = Hmat + (size_t)(row0 + col) * DEV_H2 + 2 * hi;

    v8f c[2] = {};

    for (int k0 = 0; k0 < DEV_H2; k0 += 4) {
        v2f a = *(const v2f*)(aPtr + k0);
        const float* wk = W3 + (size_t)(k0 + 2 * hi) * DEV_C;
#pragma unroll
        for (int t = 0; t < 2; ++t) {
            v2f b;
            b.x = wk[t * 16 + col];
            b.y = wk[DEV_C + t * 16 + col];
            c[t] = __builtin_amdgcn_wmma_f32_16x16x4_f32(
                false, a, false, b, (short)0, c[t], false, false);
        }
    }

    float* outBase = T3 + (size_t)row0 * DEV_C;
#pragma unroll
    for (int t = 0; t < 2; ++t)
#pragma unroll
        for (int r = 0; r < 8; ++r)
            outBase[(size_t)(r + 8 * hi) * DEV_C + t * 16 + col] = c[t][r];
}

// ---------------- edge scatter: acc[d] += t[s] * dinv[s]*dinv[d] -----------
// One thread handles 4 consecutive floats of one edge message (float4 gather,
// 4 f32 atomics). Edge ids e >= E are the implicit self-loops (s=d=e-E).
__global__ void scatter_kernel(const float* __restrict__ t, float* __restrict__ acc,
                               const float* __restrict__ dinv,
                               const int* __restrict__ src, const int* __restrict__ dst,
                               int E, int Nn, int rowStride, int colOff, int chunks) {
    long long gid = (long long)blockIdx.x * blockDim.x + threadIdx.x;
    int e  = (int)(gid / chunks);
    int c4 = (int)(gid % chunks) * 4;
    if (e >= E + Nn) return;
    int s, d;
    if (e < E) { s = src[e]; d = dst[e]; } else { s = e - E; d = s; }
    float nrm = dinv[s] * dinv[d];
    const float4 v = *(const float4*)(t + (size_t)s * rowStride + colOff + c4);
    float* p = acc + (size_t)d * rowStride + colOff + c4;
    atomicAdd(p + 0, v.x * nrm);
    atomicAdd(p + 1, v.y * nrm);
    atomicAdd(p + 2, v.z * nrm);
    atomicAdd(p + 3, v.w * nrm);
}

// ---------------- bias + relu (in place): h = relu(acc + [b1|b2]) ----------
__global__ void bias_relu_kernel(float* __restrict__ acc, const float* __restrict__ b1,
                                 const float* __restrict__ b2, long long n) {
    long long i = (long long)blockIdx.x * blockDim.x + threadIdx.x;
    if (i >= n) return;
    int cc = (int)(i & (DEV_H2 - 1));
    float bias = (cc < DEV_H) ? b1[cc] : b2[cc - DEV_H];
    float v = acc[i] + bias;
    acc[i] = v > 0.0f ? v : 0.0f;
}

// ---------------- bias + log_softmax, one wave per row (C == 32) -----------
__global__ __launch_bounds__(256)
void lsm_kernel(const float* __restrict__ acc3, const float* __restrict__ b3,
                float* __restrict__ out, int Nn) {
    const int lane = threadIdx.x & 31;
    const int row  = blockIdx.x * (blockDim.x >> 5) + (threadIdx.x >> 5);
    if (row >= Nn) return;
    float v = acc3[(size_t)row * DEV_C + lane] + b3[lane];
    float m = v;
#pragma unroll
    for (int off = 16; off; off >>= 1) m = fmaxf(m, __shfl_xor(m, off, 32));
    float ex = __expf(v - m);
    float s = ex;
#pragma unroll
    for (int off = 16; off; off >>= 1) s += __shfl_xor(s, off, 32);
    out[(size_t)row * DEV_C + lane] = v - m - __logf(s);
}

// ---------------------------------------------------------------------------
extern "C" void kernel_launch(void* const* d_in, const int* in_sizes, int n_in,
                              void* d_out, int out_size, void* d_ws, size_t ws_size,
                              hipStream_t stream) {
    const float* x  = (const float*)d_in[0];
    const int* ei1  = (const int*)d_in[1];
    const int* ei2  = (const int*)d_in[2];
    const float* W1 = (const float*)d_in[3];
    const float* b1 = (const float*)d_in[4];
    const float* W2 = (const float*)d_in[5];
    const float* b2 = (const float*)d_in[6];
    const float* W3 = (const float*)d_in[7];
    const float* b3 = (const float*)d_in[8];

    const int E1 = in_sizes[1] / 2;
    const int E2 = in_sizes[2] / 2;
    const int Nn = in_sizes[0] / DEV_D;   // 50000

    const int* src1 = ei1;           const int* dst1 = ei1 + E1;
    const int* src2 = ei2;           const int* dst2 = ei2 + E2;

    // workspace layout (floats):
    //   dinv1[N] | dinv2[N] | T[N*256] | ACC[N*256]; T3/ACC3 reuse T's space
    float* w     = (float*)d_ws;
    float* dinv1 = w;
    float* dinv2 = w + Nn;
    float* T     = w + 2 * (size_t)Nn;
    float* ACC   = T + (size_t)Nn * DEV_H2;
    float* T3    = T;                                // N*32, reuse
    float* ACC3  = T + (size_t)Nn * DEV_C;           // next N*32, disjoint
    float* out   = (float*)d_out;

    const int BLK = 256;
    const long long nAcc = (long long)Nn * DEV_H2;

    // 1) deg=1 (self-loops), zero ACC
    {
        long long total = nAcc;
        int grid = (int)((total + BLK - 1) / BLK);
        init_kernel<<<grid, BLK, 0, stream>>>(dinv1, dinv2, ACC, nAcc, Nn);
    }
    // 2) degree counts + rsqrt
    deg_kernel<<<(E1 + BLK - 1) / BLK, BLK, 0, stream>>>(dinv1, dst1, E1);
    deg_kernel<<<(E2 + BLK - 1) / BLK, BLK, 0, stream>>>(dinv2, dst2, E2);
    dinv_kernel<<<(Nn + BLK - 1) / BLK, BLK, 0, stream>>>(dinv1, Nn);
    dinv_kernel<<<(Nn + BLK - 1) / BLK, BLK, 0, stream>>>(dinv2, Nn);

    // 3) fused GEMM: T = x @ [W1 | W2]    (waves = 2 per 16-row tile)
    {
        int waves = (Nn >> 4) * 2;
        int grid  = (waves + 7) / 8;
        gemm_cat_kernel<<<grid, BLK, 0, stream>>>(x, W1, W2, T, Nn);
    }

    // 4) scatters for layer 1 and 2 (32 threads per edge, H=128)
    {
        long long th1 = (long long)(E1 + Nn) * (DEV_H / 4);
        scatter_kernel<<<(int)((th1 + BLK - 1) / BLK), BLK, 0, stream>>>(
            T, ACC, dinv1, src1, dst1, E1, Nn, DEV_H2, 0, DEV_H / 4);
        long long th2 = (long long)(E2 + Nn) * (DEV_H / 4);
        scatter_kernel<<<(int)((th2 + BLK - 1) / BLK), BLK, 0, stream>>>(
            T, ACC, dinv2, src2, dst2, E2, Nn, DEV_H2, DEV_H, DEV_H / 4);
    }

    // 5) h = relu(ACC + [b1|b2]) in place
    bias_relu_kernel<<<(int)((nAcc + BLK - 1) / BLK), BLK, 0, stream>>>(ACC, b1, b2, nAcc);

    // 6) T3 = h @ W3  (T is dead now; T3/ACC3 live in its space)
    gemm3_kernel<<<((Nn >> 4) + 7) / 8, BLK, 0, stream>>>(ACC, W3, T3, Nn);

    // 7) zero ACC3, scatter layer 3 (8 threads per edge, C=32)
    {
        long long n3 = (long long)Nn * DEV_C;
        zero_kernel<<<(int)((n3 + BLK - 1) / BLK), BLK, 0, stream>>>(ACC3, n3);
        long long th3 = (long long)(E1 + Nn) * (DEV_C / 4);
        scatter_kernel<<<(int)((th3 + BLK - 1) / BLK), BLK, 0, stream>>>(
            T3, ACC3, dinv1, src1, dst1, E1, Nn, DEV_C, 0, DEV_C / 4);
    }

    // 8) bias + log_softmax (wave per row)
    lsm_kernel<<<(Nn + 7) / 8, BLK, 0, stream>>>(ACC3, b3, out, Nn);
}